// GINE_58171037057259
// MI455X (gfx1250) — compile-verified
//
#include <hip/hip_runtime.h>
#include <hip/hip_bf16.h>

typedef __attribute__((ext_vector_type(2))) float v2f;
typedef __attribute__((ext_vector_type(4))) float v4f;
typedef __attribute__((ext_vector_type(8))) float v8f;

#define D 128
#define NPAIRS 64        // 128 K-values stored as 64 interleaved K-pairs
#define PSTRIDE 144      // float2 stride per pair-row; 144 % 32 == 16 -> the two
                         // half-waves (pair p vs p+1) hit disjoint 32-bank groups

// ---------------------------------------------------------------- utilities

__global__ void copy_f4_kernel(float* __restrict__ dst, const float* __restrict__ src, int n4) {
    int i = blockIdx.x * blockDim.x + threadIdx.x;
    if (i < n4) ((float4*)dst)[i] = ((const float4*)src)[i];
}

__global__ void zero_kernel(float* __restrict__ p, int n) {
    int i = blockIdx.x * blockDim.x + threadIdx.x;
    if (i < n) p[i] = 0.0f;
}

// ------------------------------------------------- GINE message + scatter-add
// h[dst] += relu(x[src] + edge_attr), one float4 chunk per thread.
// edge_attr is a one-pass 409MB stream -> non-temporal loads preserve L2 for
// the random x-gather (x = 25.6MB, fully L2-resident) and the atomic targets.
__global__ void scatter_kernel(float* __restrict__ h, const float* __restrict__ x,
                               const float* __restrict__ ea, const int* __restrict__ ei,
                               int E) {
    int i = blockIdx.x * blockDim.x + threadIdx.x;
    if (i >= E * 32) return;                  // 32 float4 chunks per edge (D=128)
    int e  = i >> 5;
    int cg = i & 31;
    int src = ei[e];
    int dst = ei[E + e];
    const float4 xv = ((const float4*)x)[(size_t)src * 32 + cg];
    const v4f    av = __builtin_nontemporal_load(((const v4f*)ea) + (size_t)e * 32 + cg);
    float m0 = fmaxf(xv.x + av.x, 0.0f);
    float m1 = fmaxf(xv.y + av.y, 0.0f);
    float m2 = fmaxf(xv.z + av.z, 0.0f);
    float m3 = fmaxf(xv.w + av.w, 0.0f);
    float* hp = h + (size_t)dst * D + cg * 4;
    unsafeAtomicAdd(hp + 0, m0);              // -> global_atomic_add_f32 (no return)
    unsafeAtomicAdd(hp + 1, m1);
    unsafeAtomicAdd(hp + 2, m2);
    unsafeAtomicAdd(hp + 3, m3);
}

// ------------------------------------------------------------- WMMA GEMM
// out[MxD] = act(A) @ W[DxD] + bias, act(a) = USEPRE ? relu(a*preScale+preShift) : a
// Block: 256 threads = 8 waves; each wave owns one 16-row tile, all 8 col tiles.
// W staged in LDS as K-pair-interleaved float2: Wl[p*PSTRIDE + n] = {W[2p][n], W[2p+1][n]}
// so every B fragment is a single aligned ds_load_b64 straight into an even VGPR
// pair (no v_mov repacking before v_wmma). A fragments are loaded once from
// global and reused across all 8 column tiles (8 v8f accumulators in registers).
template <int USEPRE>
__global__ __launch_bounds__(256) void gemm_wmma_kernel(
    const float* __restrict__ A, const float* __restrict__ W,
    const float* __restrict__ bias,
    const float* __restrict__ preScale, const float* __restrict__ preShift,
    float* __restrict__ out, int Mtiles) {

    __shared__ float Wl[NPAIRS * PSTRIDE * 2];   // 73,728 bytes
    // cooperative repack: pair p, col n  <-  rows 2p, 2p+1 of W (coalesced reads)
    for (int i = threadIdx.x; i < NPAIRS * D; i += 256) {
        int p = i >> 7;          // K-pair
        int n = i & 127;         // column
        float2 w2;
        w2.x = W[(2 * p)     * D + n];
        w2.y = W[(2 * p + 1) * D + n];
        *(float2*)&Wl[(p * PSTRIDE + n) * 2] = w2;
    }
    __syncthreads();

    const int wave  = threadIdx.x >> 5;
    const int lane  = threadIdx.x & 31;
    const int half  = lane >> 4;      // which half-wave
    const int r     = lane & 15;      // row within A tile / col within B tile
    const int mtile = blockIdx.x * 8 + wave;
    if (mtile >= Mtiles) return;      // whole-wave exit: EXEC all-ones for WMMA

    const int row0 = mtile * 16;
    const float* Arow = A + (size_t)(row0 + r) * D;
    __builtin_prefetch(Arow, 0, 3);   // warm this lane's 512B A-strip row

    v8f acc[8] = {};                  // 16x128 output strip, in registers

    for (int k = 0; k < D; k += 4) {
        const int ka = k + 2 * half;  // A frag: lane (half,r) holds (M=r, K=ka, ka+1)
        float a0 = Arow[ka];
        float a1 = Arow[ka + 1];
        if (USEPRE) {                 // fused BN + ReLU on GEMM input
            a0 = fmaxf(a0 * preScale[ka]     + preShift[ka],     0.0f);
            a1 = fmaxf(a1 * preScale[ka + 1] + preShift[ka + 1], 0.0f);
        }
        v2f a; a[0] = a0; a[1] = a1;
        const int prow = (k >> 1) + half;          // K-pair row for this half-wave
        const float* bbase = &Wl[(prow * PSTRIDE + r) * 2];
        #pragma unroll
        for (int ct = 0; ct < 8; ++ct) {
            const v2f b = *(const v2f*)(bbase + ct * 32);   // ds_load_b64, aligned pair
            acc[ct] = __builtin_amdgcn_wmma_f32_16x16x4_f32(
                false, a, false, b, (short)0, acc[ct], false, false);
        }
    }

    // epilogue: +bias, store. C layout: VGPR v holds M = v + 8*half, N = r.
    #pragma unroll
    for (int ct = 0; ct < 8; ++ct) {
        const int col = ct * 16 + r;
        const float bb = bias[col];
        #pragma unroll
        for (int v = 0; v < 8; ++v) {
            const int m = v + 8 * half;
            out[(size_t)(row0 + m) * D + col] = acc[ct][v] + bb;
        }
    }
}

// ----------------------------------------------- BN column statistics (sum, sumsq)
__global__ void colstats_kernel(const float* __restrict__ t, float* __restrict__ sum,
                                float* __restrict__ sumsq, int Nrows) {
    const int c = threadIdx.x;        // 128 threads = 128 columns
    float s = 0.0f, q = 0.0f;
    for (int row = blockIdx.x; row < Nrows; row += gridDim.x) {
        float v = t[(size_t)row * D + c];
        s += v; q += v * v;
    }
    unsafeAtomicAdd(&sum[c], s);
    unsafeAtomicAdd(&sumsq[c], q);
}

// fold BN into per-column affine: scale = g*rsqrt(var+eps), shift = beta - mean*scale
__global__ void bn_finalize_kernel(const float* __restrict__ sum, const float* __restrict__ sumsq,
                                   const float* __restrict__ gamma, const float* __restrict__ beta,
                                   float* __restrict__ scale, float* __restrict__ shift,
                                   float invN) {
    const int c = threadIdx.x;
    float m   = sum[c] * invN;
    float var = sumsq[c] * invN - m * m;   // biased variance, matches jnp.var
    float s   = gamma[c] * rsqrtf(var + 1e-5f);
    scale[c] = s;
    shift[c] = beta[c] - m * s;
}

// out = (relu?)(u*scale + shift), float4-vectorized
__global__ void affine_act_kernel(const float* __restrict__ u, const float* __restrict__ scale,
                                  const float* __restrict__ shift, float* __restrict__ out,
                                  int relu, int n4) {
    int i = blockIdx.x * blockDim.x + threadIdx.x;
    if (i >= n4) return;
    const int cg = i & 31;                 // D/4 = 32 float4 columns
    const float4 uu = ((const float4*)u)[i];
    const float4 sc = ((const float4*)scale)[cg];
    const float4 sh = ((const float4*)shift)[cg];
    float4 o;
    o.x = uu.x * sc.x + sh.x;
    o.y = uu.y * sc.y + sh.y;
    o.z = uu.z * sc.z + sh.z;
    o.w = uu.w * sc.w + sh.w;
    if (relu) {
        o.x = fmaxf(o.x, 0.0f); o.y = fmaxf(o.y, 0.0f);
        o.z = fmaxf(o.z, 0.0f); o.w = fmaxf(o.w, 0.0f);
    }
    ((float4*)out)[i] = o;
}

// ---------------------------------------------------------------- launcher

extern "C" void kernel_launch(void* const* d_in, const int* in_sizes, int n_in,
                              void* d_out, int out_size, void* d_ws, size_t ws_size,
                              hipStream_t stream) {
    const float* x_in  = (const float*)d_in[0];
    const int*   ei    = (const int*)d_in[1];
    const float* ea    = (const float*)d_in[2];
    const float* W1    = (const float*)d_in[4];
    const float* b1    = (const float*)d_in[5];
    const float* g_mlp = (const float*)d_in[6];
    const float* be_mlp= (const float*)d_in[7];
    const float* W2    = (const float*)d_in[8];
    const float* b2    = (const float*)d_in[9];
    const float* g_out = (const float*)d_in[10];
    const float* be_out= (const float*)d_in[11];

    const int N  = in_sizes[0] / D;        // 50000
    const int E  = in_sizes[1] / 2;        // 800000
    const int ND = N * D;
    const int Mtiles = N / 16;             // 3125 (exact)
    const int L = 3;

    float* ws    = (float*)d_ws;
    float* h     = ws;                     // GEMM1 input (x + aggr)
    float* t     = h  + ND;                // GEMM1 output
    float* u     = t  + ND;                // GEMM2 output
    float* xb    = u  + ND;                // layer output (layers 0,1)
    float* sum   = xb + ND;
    float* sumsq = sum + D;
    float* scale = sumsq + D;
    float* shift = scale + D;

    const dim3 blk(256);
    const dim3 gCopy((ND / 4 + 255) / 256);
    const dim3 gScat((E * 32 + 255) / 256);
    const dim3 gGemm((Mtiles + 7) / 8);
    const dim3 gStat(256);
    const float invN = 1.0f / (float)N;

    const float* xcur = x_in;
    for (int i = 0; i < L; ++i) {
        const float* W1i = W1 + (size_t)i * D * D;
        const float* W2i = W2 + (size_t)i * D * D;

        // h = x + scatter_add(relu(x[src] + edge_attr))
        copy_f4_kernel<<<gCopy, blk, 0, stream>>>(h, xcur, ND / 4);
        scatter_kernel<<<gScat, blk, 0, stream>>>(h, xcur, ea, ei, E);

        // t = h @ W1 + b1
        zero_kernel<<<1, 256, 0, stream>>>(sum, 2 * D);
        gemm_wmma_kernel<0><<<gGemm, blk, 0, stream>>>(h, W1i, b1 + i * D,
                                                       nullptr, nullptr, t, Mtiles);
        // BN_mlp stats -> (scale, shift)
        colstats_kernel<<<gStat, dim3(D), 0, stream>>>(t, sum, sumsq, N);
        bn_finalize_kernel<<<1, D, 0, stream>>>(sum, sumsq, g_mlp + i * D, be_mlp + i * D,
                                                scale, shift, invN);
        // u = relu(bn(t)) @ W2 + b2   (BN+ReLU fused into GEMM A-load)
        gemm_wmma_kernel<1><<<gGemm, blk, 0, stream>>>(t, W2i, b2 + i * D,
                                                       scale, shift, u, Mtiles);
        // BN_out stats -> (scale, shift)
        zero_kernel<<<1, 256, 0, stream>>>(sum, 2 * D);
        colstats_kernel<<<gStat, dim3(D), 0, stream>>>(u, sum, sumsq, N);
        bn_finalize_kernel<<<1, D, 0, stream>>>(sum, sumsq, g_out + i * D, be_out + i * D,
                                                scale, shift, invN);
        // x_next = (relu if i<L-1)(bn(u))
        float* dst = (i == L - 1) ? (float*)d_out : xb;
        affine_act_kernel<<<gCopy, blk, 0, stream>>>(u, scale, shift, dst,
                                                     (i < L - 1) ? 1 : 0, ND / 4);
        xcur = xb;
    }
}